// CausalDepthwiseConv1d_28338194219158
// MI455X (gfx1250) — compile-verified
//
#include <hip/hip_runtime.h>

typedef float v2f __attribute__((ext_vector_type(2)));
typedef float v8f __attribute__((ext_vector_type(8)));

#define B_ 8
#define L_ 4096
#define C_ 512
#define K_ 5
#define S_ 8
#define MSUB 4        // 16-row WMMA M-subtiles per block
#define MTILE (MSUB * 16)   // 64 rows per block
#define ASTRIDE 516   // 512 + 4 floats pad: 516 % 64 == 4 -> conflict-free b64 reads

// ---------------------------------------------------------------------------
// Fused kernel: per-segment causal depthwise conv (K=5) fused into the A-tile
// fill of the pointwise GEMM  out[m,n] = sum_c dw[m,c] * w_pw[n,c] + b_pw[n],
// M = B*L = 32768, N = K = 512.  Full fp32 via V_WMMA_F32_16X16X4_F32.
//
// Block: 256 threads = 8 waves; one 64-row M-tile per block (64 consecutive
// l positions of one batch; 4096 % 64 == 0 so no batch straddle).  Each wave
// owns 4 N-tiles of 16 (8 waves x 64 = 512 = N) and all FOUR 16-row
// M-subtiles, so every B fragment loaded from L2 feeds four WMMAs
// (4:1 wmma:vmem), halving w_pw L2 traffic vs the 32-row version.
//
// Segments contiguously partition [0,L) per batch, so "same segment as l" is
// exactly (l-d >= seg_start(l)); covered == true everywhere.
//
// Fragment layouts (ISA 7.12.2, 32-bit):
//   A (16x4):  lanes 0-15 hold row M=lane&15, VGPR{0,1} = K{0,1};
//              lanes 16-31 same rows, VGPR{0,1} = K{2,3}.
//   B (4x16):  mirror of A with N=lane&15.
//   C/D:       VGPR v -> M = v + 8*(lane>>4), N = lane&15.
// ---------------------------------------------------------------------------
__global__ __launch_bounds__(256) void fused_kernel(const float* __restrict__ x,
                                                    const int*   __restrict__ segb,
                                                    const float* __restrict__ w_dw,
                                                    const float* __restrict__ b_dw,
                                                    const float* __restrict__ w_pw,
                                                    const float* __restrict__ b_pw,
                                                    float*       __restrict__ out) {
  __shared__ float Atile[MTILE * ASTRIDE];   // 64 x 516 x 4B = 132 KB (<=320 KB/WGP)
  __shared__ int   sstart[MTILE];

  const int tid = threadIdx.x;
  const int m0  = blockIdx.x * MTILE;
  const int b   = m0 >> 12;          // / L_
  const int l0  = m0 & (L_ - 1);

  // --- per-row segment starts ----------------------------------------------
  if (tid < MTILE) {
    const int l = l0 + tid;
    int st = 0;
#pragma unroll
    for (int s = 0; s < S_; ++s) {
      const int a = segb[(b * S_ + s) * 2 + 0];
      const int e = segb[(b * S_ + s) * 2 + 1];
      if (l >= a && l < e) st = a;
    }
    sstart[tid] = st;
  }
  __syncthreads();

  // --- fused depthwise-conv A-tile fill ------------------------------------
  // Thread owns channels c..c+3 (fixed -> weights in registers) and 32
  // consecutive rows; rolling 5-deep float4 window over x.
  {
    const int k4    = tid & 127;
    const int c     = k4 << 2;
    const int rbase = (tid >> 7) << 5;   // 0 or 32

    float4 wt[K_];
#pragma unroll
    for (int tap = 0; tap < K_; ++tap) {
      wt[tap].x = w_dw[(c + 0) * K_ + tap];
      wt[tap].y = w_dw[(c + 1) * K_ + tap];
      wt[tap].z = w_dw[(c + 2) * K_ + tap];
      wt[tap].w = w_dw[(c + 3) * K_ + tap];
    }
    const float4 bias = *(const float4*)&b_dw[c];

    const float* xb = x + ((size_t)b * L_) * C_ + c;
    const int lstart = l0 + rbase;

    // win[j] holds x at position (l - 4 + j) for the current row l.
    float4 win[K_];
#pragma unroll
    for (int j = 0; j < K_; ++j) {
      const int lp = lstart - 4 + j;
      if (lp >= 0) win[j] = *(const float4*)&xb[(size_t)lp * C_];
      else         win[j] = make_float4(0.f, 0.f, 0.f, 0.f);
    }

    for (int i = 0; i < 32; ++i) {
      const int r     = rbase + i;
      const int l     = lstart + i;
      const int start = sstart[r];
      float4 acc = bias;
#pragma unroll
      for (int j = 0; j < K_; ++j) {     // tap index j == K-1-d, lp = l-4+j
        if (l - 4 + j >= start) {
          acc.x += wt[j].x * win[j].x;
          acc.y += wt[j].y * win[j].y;
          acc.z += wt[j].z * win[j].z;
          acc.w += wt[j].w * win[j].w;
        }
      }
      *(float4*)&Atile[r * ASTRIDE + c] = acc;
      if (i < 31) {
#pragma unroll
        for (int j = 0; j < K_ - 1; ++j) win[j] = win[j + 1];
        win[K_ - 1] = *(const float4*)&xb[(size_t)(l + 1) * C_];
      }
    }
  }
  __syncthreads();

  // --- WMMA GEMM phase ------------------------------------------------------
  const int wave  = tid >> 5;
  const int lane  = tid & 31;
  const int half  = lane >> 4;
  const int sub   = lane & 15;         // A row / B col / D col index
  const int nbase = wave * 64;

  v8f acc[MSUB][4];                    // [m-subtile][n-tile]
#pragma unroll
  for (int t = 0; t < 4; ++t) {
    const float bp = b_pw[nbase + t * 16 + sub];
#pragma unroll
    for (int mt = 0; mt < MSUB; ++mt)
#pragma unroll
      for (int v = 0; v < 8; ++v) acc[mt][t][v] = bp;   // bias rides in C
  }

  const float* arow[MSUB];
#pragma unroll
  for (int mt = 0; mt < MSUB; ++mt)
    arow[mt] = &Atile[(sub + 16 * mt) * ASTRIDE + 2 * half];
  const float* brow = &w_pw[(size_t)(nbase + sub) * C_ + 2 * half];

  for (int kk = 0; kk < C_; kk += 4) {
    v2f a[MSUB];
#pragma unroll
    for (int mt = 0; mt < MSUB; ++mt)
      a[mt] = *(const v2f*)&arow[mt][kk];            // ds_load_b64, conflict-free
#pragma unroll
    for (int t = 0; t < 4; ++t) {
      const v2f bfrag = *(const v2f*)&brow[(size_t)t * 16 * C_ + kk];
#pragma unroll
      for (int mt = 0; mt < MSUB; ++mt) {
        acc[mt][t] = __builtin_amdgcn_wmma_f32_16x16x4_f32(
            false, a[mt], false, bfrag, (short)0, acc[mt][t], false, false);
      }
    }
  }

#pragma unroll
  for (int mt = 0; mt < MSUB; ++mt) {
#pragma unroll
    for (int t = 0; t < 4; ++t) {
      const int n = nbase + t * 16 + sub;
#pragma unroll
      for (int v = 0; v < 8; ++v) {
        const int m = mt * 16 + v + 8 * half;
        out[(size_t)(m0 + m) * C_ + n] = acc[mt][t][v];
      }
    }
  }
}

// ---------------------------------------------------------------------------
extern "C" void kernel_launch(void* const* d_in, const int* in_sizes, int n_in,
                              void* d_out, int out_size, void* d_ws, size_t ws_size,
                              hipStream_t stream) {
  (void)in_sizes; (void)n_in; (void)out_size; (void)d_ws; (void)ws_size;
  const float* x    = (const float*)d_in[0];
  const int*   segb = (const int*)  d_in[1];   // [B,S,2] int32 (JAX x64 disabled)
  const float* w_dw = (const float*)d_in[2];
  const float* b_dw = (const float*)d_in[3];
  const float* w_pw = (const float*)d_in[4];
  const float* b_pw = (const float*)d_in[5];
  float*       out  = (float*)d_out;

  fused_kernel<<<(B_ * L_) / MTILE, 256, 0, stream>>>(x, segb, w_dw, b_dw, w_pw, b_pw, out);
}